// Critique_16956530885241
// MI455X (gfx1250) — compile-verified
//
#include <hip/hip_runtime.h>

#define Hdim 1024
#define Wdim 1024
#define Npts 128
#define TILES_PER_ROW (Wdim / 16)          // 64
#define NUM_TILES (Hdim * TILES_PER_ROW)   // 65536 tiles per channel

typedef __attribute__((ext_vector_type(2))) float v2f;
typedef __attribute__((ext_vector_type(8))) float v8f;

// r = min(max(x, a), b) in one VOP3 instruction, no operand canonicalization
// (WMMA outputs are arithmetic results, never sNaN).
static __device__ __forceinline__ float maxmin(float x, float a, float b) {
    float r;
    asm("v_maxmin_num_f32 %0, %1, %2, %3" : "=v"(r) : "v"(x), "v"(a), "v"(b));
    return r;
}

// D = A(16x4) x B(4x16):
//   A row m of group g carries plane n(m,g) = (m<8) ? 8g+m : 64+8g+(m-8)
//     A row = [nv0_n, nv1_n, K_n, 0], K_n = c*nv2 - dot(x0,nv)
//   B col p = [i, j_p, 1, 0]
// C/D layout: lane L holds pixel p = L%16; VGPR r holds row m = r (+8 if L>=16).
// => low-half lanes see n = 8g+r  (across g: n = 0..63 in order),
//    high-half lanes see n = 64+8g+r (n = 64..127 in order).
// The alternating max/min fold is tracked as a composed clamp (lo,hi); each
// consecutive (even,odd) n-pair updates each register with one v_maxmin_num_f32.
// One cross-lane stitch at the end composes [0,64) with [64,128).
__global__ __launch_bounds__(256)
void plane_fold_wmma(const float* __restrict__ bp,
                     const float* __restrict__ nv,
                     float* __restrict__ out)
{
    const int c    = blockIdx.y;
    const int tid  = threadIdx.x;
    const int lane = tid & 31;
    const int wid  = tid >> 5;
    const int la   = lane & 15;
    const bool hiHalf = lane >= 16;

    __shared__ float sA[Npts];  // nv0
    __shared__ float sB[Npts];  // nv1
    __shared__ float sK[Npts];  // c*nv2 - dot(x0, nv)

    if (tid < Npts) {
        float x0 = bp[3*tid+0], x1 = bp[3*tid+1], x2 = bp[3*tid+2];
        float n0 = nv[3*tid+0], n1 = nv[3*tid+1], n2 = nv[3*tid+2];
        sA[tid] = n0;
        sB[tid] = n1;
        sK[tid] = (float)c * n2 - (x0*n0 + x1*n1 + x2*n2);
    }
    __syncthreads();

    // Tile-invariant A operands: one v2f per group.
    // A 16x4 f32 layout: lanes 0-15 hold K-cols {0,1}; lanes 16-31 K-cols {2,3}.
    v2f Aop[8];
    #pragma unroll
    for (int g = 0; g < 8; ++g) {
        int n = (la < 8) ? (8 * g + la) : (56 + 8 * g + la);  // la>=8: 64+8g+(la-8)
        Aop[g].x = hiHalf ? sK[n] : sA[n];
        Aop[g].y = hiHalf ? 0.0f  : sB[n];
    }

    const int gwave  = blockIdx.x * (blockDim.x >> 5) + wid;
    const int nwaves = gridDim.x  * (blockDim.x >> 5);
    const float INF  = __builtin_inff();

    for (int tile = gwave; tile < NUM_TILES; tile += nwaves) {
        const int i  = tile >> 6;                         // row
        const int j0 = (tile & (TILES_PER_ROW - 1)) << 4; // first col of tile

        // B 4x16 f32 layout: lanes 0-15 hold K-rows {0,1}; lanes 16-31 rows {2,3}.
        v2f Bop;
        Bop.x = hiHalf ? 1.0f : (float)i;
        Bop.y = hiHalf ? 0.0f : (float)(j0 + la);

        // Running clamp for this lane's half of the n-sequence.
        float rlo = -INF, rhi = INF;

        #pragma unroll
        for (int g = 0; g < 8; ++g) {
            v8f cz = {};
            v8f D = __builtin_amdgcn_wmma_f32_16x16x4_f32(
                        false, Aop[g], false, Bop, (short)0, cz, false, false);

            // 8 consecutive n's; parity(n) == parity(r):
            // even n -> max step, odd n -> min step => one maxmin per pair/reg.
            #pragma unroll
            for (int r = 0; r < 8; r += 2) {
                rlo = maxmin(rlo, D[r], D[r + 1]);
                rhi = maxmin(rhi, D[r], D[r + 1]);
            }
        }

        // Stitch: F = clamp over n=0..63 (low lanes), S = clamp over n=64..127
        // (high lanes). result = S(F(-inf)) = min(max(F.lo, S.lo), S.hi).
        float Flo = __shfl(rlo, la,      32);
        float Slo = __shfl(rlo, la + 16, 32);
        float Shi = __shfl(rhi, la + 16, 32);
        float res = maxmin(Flo, Slo, Shi);

        if (!hiHalf) {
            out[(size_t)c * (Hdim * Wdim) + (size_t)i * Wdim + (j0 + la)] = res;
        }
    }
}

extern "C" void kernel_launch(void* const* d_in, const int* in_sizes, int n_in,
                              void* d_out, int out_size, void* d_ws, size_t ws_size,
                              hipStream_t stream) {
    const float* bp = (const float*)d_in[0];   // basepoints [128,3]
    const float* nv = (const float*)d_in[1];   // normal_vectors [128,3]
    float* out = (float*)d_out;                // [3,1024,1024]
    dim3 grid(2048, 3, 1);                     // 16384 waves/channel-slice, 4 tiles each
    plane_fold_wmma<<<grid, 256, 0, stream>>>(bp, nv, out);
}